// Graph_Physics_Attention_1D_15599321219566
// MI455X (gfx1250) — compile-verified
//
#include <hip/hip_runtime.h>
#include <math.h>

// ---------------------------------------------------------------------------
// Graph Physics Attention 1D — CDNA5 (gfx1250).
// Matrix math on V_WMMA_F32_16X16X4_F32 (fp32-exact); GEMM staging uses
// double-buffered async global->LDS b128 copies (ASYNCcnt-tracked).
// ---------------------------------------------------------------------------

typedef __attribute__((ext_vector_type(2))) float v2f;
typedef __attribute__((ext_vector_type(8))) float v8f;
typedef __attribute__((ext_vector_type(4))) int   v4i;

static constexpr int N_NODES = 30000;
static constexpr int DIM     = 256;
static constexpr int H       = 8;
static constexpr int D       = 64;
static constexpr int S       = 32;
static constexpr int B       = 8;
static constexpr int INNER   = H * D;   // 512

#ifndef __has_builtin
#define __has_builtin(x) 0
#endif
#if __has_builtin(__builtin_amdgcn_global_load_async_to_lds_b128)
#define HAVE_ASYNC_LDS 1
#else
#define HAVE_ASYNC_LDS 0
#endif

typedef __attribute__((address_space(1))) v4i gv4i_t;   // global int4*
typedef __attribute__((address_space(3))) v4i lv4i_t;   // LDS int4*

__device__ __forceinline__ v8f wmma_f32_4(v2f a, v2f b, v8f c) {
  // D(16x16,f32) = A(16x4,f32) * B(4x16,f32) + C   [wave32]
  return __builtin_amdgcn_wmma_f32_16x16x4_f32(
      false, a, false, b, (short)0, c, false, false);
}

__device__ __forceinline__ float4 f4zero() { return make_float4(0.f, 0.f, 0.f, 0.f); }

// 16-byte global -> LDS stage; async (ASYNCcnt-tracked) when available.
__device__ __forceinline__ void stage_f4(float* dst_lds, const float* src) {
#if HAVE_ASYNC_LDS
  __builtin_amdgcn_global_load_async_to_lds_b128(
      (gv4i_t*)src, (lv4i_t*)dst_lds, 0, 0);
#else
  *(float4*)dst_lds = *(const float4*)src;
#endif
}

__device__ __forceinline__ void wait_async_all() {
#if HAVE_ASYNC_LDS
#if __has_builtin(__builtin_amdgcn_s_wait_asynccnt)
  __builtin_amdgcn_s_wait_asynccnt(0);
#else
  asm volatile("s_wait_asynccnt 0x0" ::: "memory");
#endif
#endif
}

// ---------------------------------------------------------------------------
// GEMM: Y[nrows, NO] = X[nrows, K] @ W[K, NO] + bias[NO].  K, NO compile-time.
// 256 threads = 8 waves (4 row-groups x 2 col-groups); block tile 64x128;
// wave tile 16x64 (4 accumulators -> 1.25 LDS ops per WMMA); K chunks of 32,
// double-buffered through LDS via async b128 copies.
// ---------------------------------------------------------------------------
template <int K, int NO>
__global__ __launch_bounds__(256) void gemm_wmma_kernel(
    const float* __restrict__ X, const float* __restrict__ W,
    const float* __restrict__ bias, float* __restrict__ Y, int nrows) {
  constexpr int KC  = 32;
  constexpr int NCH = K / KC;
  __shared__ float xs[2][64][36];     // [buf][row][k]   (36: 16B-aligned rows)
  __shared__ float wsh[2][32][132];   // [buf][k][col]   (132: 16B-aligned rows)

  const int tid  = threadIdx.x;
  const int wid  = tid >> 5;
  const int lane = tid & 31;
  const int lo   = lane & 15;         // A row (M) / B,C col (N)
  const int hi   = lane >> 4;         // K half-select
  const int rw   = wid >> 1;          // row group 0..3
  const int cg   = wid & 1;           // col group 0..1
  const int colBase = blockIdx.y * 128;
  const int row0    = blockIdx.x * 64;

  auto stage = [&](int buf, int kc) {
    #pragma unroll
    for (int i = 0; i < 2; ++i) {     // X tile: 64 rows x 32 k
      int e = tid + 256 * i;
      int r = e >> 3, c4 = (e & 7) * 4;
      int gr = min(row0 + r, nrows - 1);          // clamp: A-row garbage is
      stage_f4(&xs[buf][r][c4], &X[(size_t)gr * K + kc + c4]);  // never stored
    }
    #pragma unroll
    for (int i = 0; i < 4; ++i) {     // W tile: 32 k x 128 cols
      int e = tid + 256 * i;
      int r = e >> 5, c4 = (e & 31) * 4;
      stage_f4(&wsh[buf][r][c4], &W[(size_t)(kc + r) * NO + colBase + c4]);
    }
  };

  v8f acc[4] = {};
  stage(0, 0);
  for (int ci = 0; ci < NCH; ++ci) {
    wait_async_all();
    __syncthreads();                  // chunk ci resident in LDS for all waves
    if (ci + 1 < NCH) stage((ci + 1) & 1, (ci + 1) * KC);
    const int buf = ci & 1;
    #pragma unroll
    for (int kk = 0; kk < KC; kk += 4) {
      v2f a = *(const v2f*)&xs[buf][rw * 16 + lo][kk + 2 * hi];  // ds_load_b64
      #pragma unroll
      for (int j = 0; j < 4; ++j) {
        v2f bb;
        bb[0] = wsh[buf][kk + 2 * hi][cg * 64 + j * 16 + lo];
        bb[1] = wsh[buf][kk + 2 * hi + 1][cg * 64 + j * 16 + lo];
        acc[j] = wmma_f32_4(a, bb, acc[j]);
      }
    }
    __syncthreads();                  // everyone done with buf before rewrite
  }

  const int colW = colBase + cg * 64;
  #pragma unroll
  for (int j = 0; j < 4; ++j) {
    const float bv = bias[colW + j * 16 + lo];
    #pragma unroll
    for (int r = 0; r < 8; ++r) {
      int row = row0 + rw * 16 + r + 8 * hi;      // C layout: M = r + 8*hi
      if (row < nrows) Y[(size_t)row * NO + colW + j * 16 + lo] = acc[j][r] + bv;
    }
  }
}

// ---------------------------------------------------------------------------
// Slice logits: w[n,h,s] = (xm[n,h,:] @ W_slice + b_slice) / temp[h].
// ---------------------------------------------------------------------------
__global__ void slice_logits_kernel(const float* __restrict__ xm,
                                    const float* __restrict__ Wsl,
                                    const float* __restrict__ bsl,
                                    const float* __restrict__ temp,
                                    float* __restrict__ wout, int n) {
  __shared__ float wsl[64][36];
  const int tid = threadIdx.x;
  #pragma unroll
  for (int i = 0; i < 2; ++i) {                  // 64x32 floats
    int e = tid + 256 * i;
    int r = e >> 3, c4 = (e & 7) * 4;
    *(float4*)&wsl[r][c4] = *(const float4*)&Wsl[r * S + c4];
  }
  __syncthreads();

  const int h    = blockIdx.y;
  const int wid  = tid >> 5;
  const int lane = tid & 31;
  const int lo = lane & 15, hi = lane >> 4;
  const int sTile = wid & 1;
  const int n0 = blockIdx.x * 64 + (wid >> 1) * 16;
  const int rowA = min(n0 + lo, n - 1);          // clamp: garbage never stored

  v8f acc = {};
  #pragma unroll
  for (int kk = 0; kk < 64; kk += 4) {
    v2f a = *(const v2f*)&xm[(size_t)rowA * INNER + h * D + kk + 2 * hi];
    v2f bb;
    bb[0] = wsl[kk + 2 * hi][sTile * 16 + lo];
    bb[1] = wsl[kk + 2 * hi + 1][sTile * 16 + lo];
    acc = wmma_f32_4(a, bb, acc);
  }

  const float tinv = 1.f / temp[h];
  const int sc = sTile * 16 + lo;
  const float bv = bsl[sc];
  #pragma unroll
  for (int r = 0; r < 8; ++r) {
    int row = n0 + r + 8 * hi;
    if (row < n) wout[(size_t)row * (H * S) + h * S + sc] = (acc[r] + bv) * tinv;
  }
}

// Softmax over S=32 per (n,h) row, in place, float4-vectorized.
__global__ void softmax_s_kernel(float* __restrict__ w, int total_rows) {
  int row = blockIdx.x * blockDim.x + threadIdx.x;
  if (row >= total_rows) return;
  float4* p = (float4*)(w + (size_t)row * S);
  float4 v[8];
  #pragma unroll
  for (int i = 0; i < 8; ++i) v[i] = p[i];
  float m = -INFINITY;
  #pragma unroll
  for (int i = 0; i < 8; ++i)
    m = fmaxf(m, fmaxf(fmaxf(v[i].x, v[i].y), fmaxf(v[i].z, v[i].w)));
  float sum = 0.f;
  #pragma unroll
  for (int i = 0; i < 8; ++i) {
    v[i].x = __expf(v[i].x - m); v[i].y = __expf(v[i].y - m);
    v[i].z = __expf(v[i].z - m); v[i].w = __expf(v[i].w - m);
    sum += v[i].x + v[i].y + v[i].z + v[i].w;
  }
  float inv = 1.f / sum;
  #pragma unroll
  for (int i = 0; i < 8; ++i) {
    v[i].x *= inv; v[i].y *= inv; v[i].z *= inv; v[i].w *= inv;
    p[i] = v[i];
  }
}

// Segment starts of sorted batch: seg[b] = lower_bound(batch, b); seg[B] = n.
__global__ void seg_kernel(const int* __restrict__ batch, int n, int* __restrict__ seg) {
  int b = threadIdx.x;
  if (b > B) return;
  if (b == B) { seg[B] = n; return; }
  int lo = 0, hi = n;
  while (lo < hi) { int mid = (lo + hi) >> 1; if (batch[mid] < b) lo = mid + 1; else hi = mid; }
  seg[b] = lo;
}

// slice_norm[b,h,s] += sum_n w[n,h,s]; sorted batch -> uniform register flush.
__global__ void slice_norm_kernel(const float* __restrict__ w,
                                  const int* __restrict__ batch,
                                  float* __restrict__ slice_norm, int n) {
  const int t  = threadIdx.x;                    // (h,s): H*S == 256
  const int n0 = blockIdx.x * 512;
  const int n1 = min(n0 + 512, n);
  if (n0 >= n1) return;
  int cur = batch[n0];
  float acc = 0.f;
  for (int i = n0; i < n1; ++i) {
    int b = batch[i];                            // uniform across block
    if (b != cur) {
      unsafeAtomicAdd(&slice_norm[cur * (H * S) + t], acc);
      acc = 0.f; cur = b;
    }
    acc += w[(size_t)i * (H * S) + t];
  }
  unsafeAtomicAdd(&slice_norm[cur * (H * S) + t], acc);
}

// ---------------------------------------------------------------------------
// Scatter-pool GEMM: slice_token[b,h] = w[range,h,:]^T @ fx[range,h,:]
// Split-K over blockIdx.y; 64-node chunks staged (zero-padded) in LDS so the
// inner loop is a uniform 16-WMMA unroll; global atomic f32 accumulation.
// ---------------------------------------------------------------------------
__global__ void scatter_pool_kernel(const float* __restrict__ w,
                                    const float* __restrict__ fx,
                                    const int* __restrict__ seg,
                                    float* __restrict__ slice_token) {
  __shared__ float wt[64][36];   // [node][s]
  __shared__ float ft[64][68];   // [node][d]
  const int bh = blockIdx.x;
  const int b = bh >> 3, h = bh & 7;
  const int tid = threadIdx.x;
  const int wid = tid >> 5, lane = tid & 31;
  const int lo = lane & 15, hi = lane >> 4;
  const int sTile = wid >> 2;            // 0..1
  const int dTile = wid & 3;             // 0..3
  const int kstart = seg[b], kend = seg[b + 1];
  const int per = (kend - kstart + (int)gridDim.y - 1) / (int)gridDim.y;
  const int k0 = kstart + (int)blockIdx.y * per;
  const int k1 = min(k0 + per, kend);

  v8f acc = {};
  for (int kb = k0; kb < k1; kb += 64) {
    #pragma unroll
    for (int i = 0; i < 2; ++i) {        // w rows: 64x32
      int e = tid + 256 * i;
      int r = e >> 3, c4 = (e & 7) * 4;
      int node = kb + r;
      float4 v = (node < k1) ? *(const float4*)&w[(size_t)node * (H * S) + h * S + c4] : f4zero();
      *(float4*)&wt[r][c4] = v;
    }
    #pragma unroll
    for (int i = 0; i < 4; ++i) {        // fx rows: 64x64
      int e = tid + 256 * i;
      int r = e >> 4, c4 = (e & 15) * 4;
      int node = kb + r;
      float4 v = (node < k1) ? *(const float4*)&fx[(size_t)node * INNER + h * D + c4] : f4zero();
      *(float4*)&ft[r][c4] = v;
    }
    __syncthreads();
    #pragma unroll
    for (int kk = 0; kk < 64; kk += 4) {
      int nl = kk + 2 * hi;
      v2f a, bb;
      a[0]  = wt[nl][sTile * 16 + lo];
      a[1]  = wt[nl + 1][sTile * 16 + lo];
      bb[0] = ft[nl][dTile * 16 + lo];
      bb[1] = ft[nl + 1][dTile * 16 + lo];
      acc = wmma_f32_4(a, bb, acc);
    }
    __syncthreads();
  }
  #pragma unroll
  for (int r = 0; r < 8; ++r) {
    int s = sTile * 16 + r + 8 * hi;
    unsafeAtomicAdd(&slice_token[(((size_t)b * H + h) * S + s) * D + dTile * 16 + lo], acc[r]);
  }
}

// ---------------------------------------------------------------------------
// Per-(b,h) slice attention, LDS-resident (32 tokens x 64 dims). Tiny.
// ---------------------------------------------------------------------------
__global__ void slice_attn_kernel(const float* __restrict__ slice_token,
                                  const float* __restrict__ slice_norm,
                                  const float* __restrict__ Wq,
                                  const float* __restrict__ Wk,
                                  const float* __restrict__ Wv,
                                  float* __restrict__ out_tok) {
  __shared__ float st[S][D], qm[S][D], km[S][D], vm[S][D], at[S][S + 1];
  const int bh = blockIdx.x;
  const int t  = threadIdx.x;
  const size_t base = (size_t)bh * S * D;

  for (int e = t; e < S * D; e += 256) {
    int s = e >> 6;
    st[s][e & 63] = slice_token[base + e] / (slice_norm[bh * S + s] + 1e-5f);
  }
  __syncthreads();
  for (int e = t; e < S * D; e += 256) {
    int s = e >> 6, d = e & 63;
    float aq = 0.f, ak = 0.f, av = 0.f;
    #pragma unroll 8
    for (int j = 0; j < D; ++j) {
      float x = st[s][j];
      aq += x * Wq[j * D + d]; ak += x * Wk[j * D + d]; av += x * Wv[j * D + d];
    }
    qm[s][d] = aq; km[s][d] = ak; vm[s][d] = av;
  }
  __syncthreads();
  for (int e = t; e < S * S; e += 256) {
    int s = e >> 5, tt = e & 31;
    float acc = 0.f;
    #pragma unroll 8
    for (int j = 0; j < D; ++j) acc += qm[s][j] * km[tt][j];
    at[s][tt] = acc * 0.125f;               // D^-0.5
  }
  __syncthreads();
  if (t < S) {
    float m = -INFINITY;
    #pragma unroll
    for (int j = 0; j < S; ++j) m = fmaxf(m, at[t][j]);
    float sum = 0.f;
    #pragma unroll
    for (int j = 0; j < S; ++j) { float ex = __expf(at[t][j] - m); at[t][j] = ex; sum += ex; }
    float inv = 1.f / sum;
    #pragma unroll
    for (int j = 0; j < S; ++j) at[t][j] *= inv;
  }
  __syncthreads();
  for (int e = t; e < S * D; e += 256) {
    int s = e >> 6, d = e & 63;
    float acc = 0.f;
    #pragma unroll
    for (int j = 0; j < S; ++j) acc += at[s][j] * vm[j][d];
    out_tok[base + e] = acc;
  }
}

// ---------------------------------------------------------------------------
// Gather as WMMA: out_x[tile of 16 nodes, h, :] = w-frag x out_tok[b,h].
// Sorted batch: loop over the (usually 1) graph ids in the tile, masking A
// rows per-lane. w fragments preloaded once as global b64, reused across ids.
// ---------------------------------------------------------------------------
__global__ void gather_wmma_kernel(const float* __restrict__ w,
                                   const int* __restrict__ batch,
                                   const float* __restrict__ out_tok,
                                   float* __restrict__ out_x, int n) {
  const int h = blockIdx.y;
  const int wid = threadIdx.x >> 5, lane = threadIdx.x & 31;
  const int lo = lane & 15, hi = lane >> 4;
  const int dTile = wid & 3;
  const int row0 = blockIdx.x * 32 + (wid >> 2) * 16;
  if (row0 >= n) return;                       // wave-uniform

  const int myrow = min(row0 + lo, n - 1);     // clamp: extra rows never stored
  const int myb = batch[myrow];
  v2f wv[8];                                   // k = 4*t + 2*hi + {0,1}
  #pragma unroll
  for (int t2 = 0; t2 < 8; ++t2)
    wv[t2] = *(const v2f*)&w[(size_t)myrow * (H * S) + h * S + 4 * t2 + 2 * hi];

  const int bmin = batch[row0];
  const int bmax = batch[min(row0 + 15, n - 1)];

  v8f acc = {};
  for (int bb = bmin; bb <= bmax; ++bb) {
    const float* OT = out_tok + (((size_t)bb * H + h) * S) * D + dTile * 16 + lo;
    #pragma unroll
    for (int t2 = 0; t2 < 8; ++t2) {
      int k0 = 4 * t2 + 2 * hi;
      v2f a, bf;
      a[0] = (myb == bb) ? wv[t2][0] : 0.f;
      a[1] = (myb == bb) ? wv[t2][1] : 0.f;
      bf[0] = OT[k0 * D];
      bf[1] = OT[(k0 + 1) * D];
      acc = wmma_f32_4(a, bf, acc);
    }
  }
  #pragma unroll
  for (int r = 0; r < 8; ++r) {
    int row = row0 + r + 8 * hi;
    if (row < n) out_x[(size_t)row * INNER + h * D + dTile * 16 + lo] = acc[r];
  }
}

// ---------------------------------------------------------------------------
extern "C" void kernel_launch(void* const* d_in, const int* in_sizes, int n_in,
                              void* d_out, int out_size, void* d_ws, size_t ws_size,
                              hipStream_t stream) {
  const float* x       = (const float*)d_in[0];
  const int*   batch   = (const int*)  d_in[1];
  const float* W_fx    = (const float*)d_in[2];
  const float* b_fx    = (const float*)d_in[3];
  const float* W_x     = (const float*)d_in[4];
  const float* b_x     = (const float*)d_in[5];
  const float* W_slice = (const float*)d_in[6];
  const float* b_slice = (const float*)d_in[7];
  const float* g_temp  = (const float*)d_in[8];
  const float* W_q     = (const float*)d_in[9];
  const float* W_k     = (const float*)d_in[10];
  const float* W_v     = (const float*)d_in[11];
  const float* W_out   = (const float*)d_in[12];
  const float* b_out   = (const float*)d_in[13];
  float* out = (float*)d_out;

  char* ws = (char*)d_ws;
  const size_t SZ_FX = (size_t)N_NODES * INNER * sizeof(float);
  const size_t SZ_W  = (size_t)N_NODES * H * S * sizeof(float);
  float* fx         = (float*)(ws);
  float* xm         = (float*)(ws + SZ_FX);                 // reused as out_x
  float* wbuf       = (float*)(ws + 2 * SZ_FX);
  float* slice_norm = (float*)(ws + 2 * SZ_FX + SZ_W);
  float* slice_tok  = slice_norm + (size_t)B * H * S;
  float* out_tok    = slice_tok + (size_t)B * H * S * D;
  int*   seg        = (int*)(out_tok + (size_t)B * H * S * D);
  float* out_x      = xm;

  (void)hipMemsetAsync(slice_norm, 0,
                       ((size_t)B * H * S + (size_t)B * H * S * D) * sizeof(float), stream);

  const int NT32 = (N_NODES + 31) / 32;    // 938
  const int NT64 = (N_NODES + 63) / 64;    // 469

  gemm_wmma_kernel<DIM, INNER><<<dim3(NT64, INNER / 128), 256, 0, stream>>>(x, W_fx, b_fx, fx, N_NODES);
  gemm_wmma_kernel<DIM, INNER><<<dim3(NT64, INNER / 128), 256, 0, stream>>>(x, W_x,  b_x,  xm, N_NODES);

  slice_logits_kernel<<<dim3(NT64, H), 256, 0, stream>>>(xm, W_slice, b_slice, g_temp, wbuf, N_NODES);
  softmax_s_kernel<<<(N_NODES * H + 255) / 256, 256, 0, stream>>>(wbuf, N_NODES * H);

  seg_kernel<<<1, 32, 0, stream>>>(batch, N_NODES, seg);
  slice_norm_kernel<<<(N_NODES + 511) / 512, 256, 0, stream>>>(wbuf, batch, slice_norm, N_NODES);
  scatter_pool_kernel<<<dim3(B * H, 16), 256, 0, stream>>>(wbuf, fx, seg, slice_tok);

  slice_attn_kernel<<<B * H, 256, 0, stream>>>(slice_tok, slice_norm, W_q, W_k, W_v, out_tok);

  gather_wmma_kernel<<<dim3(NT32, H), 256, 0, stream>>>(wbuf, batch, out_tok, out_x, N_NODES);
  gemm_wmma_kernel<INNER, DIM><<<dim3(NT64, DIM / 128), 256, 0, stream>>>(out_x, W_out, b_out, out, N_NODES);

  (void)in_sizes; (void)n_in; (void)out_size; (void)ws_size;
}